// TemporalShift_44555990729319
// MI455X (gfx1250) — compile-verified
//
#include <hip/hip_runtime.h>

typedef __attribute__((ext_vector_type(2))) float v2f;
typedef __attribute__((ext_vector_type(8))) float v8f;

#define SEQ_C 768
#define ROW_STRIDE 151296   /* 197*768 */
#define NROWS 512
#define LN_EPS 1e-5f

// ---------------- bulk passthrough copy: x -> out ----------------
__global__ __launch_bounds__(256) void copy_kernel(const float4* __restrict__ in,
                                                   float4* __restrict__ out, int n4) {
  int idx = blockIdx.x * 256 + threadIdx.x;
  if (idx < n4) out[idx] = in[idx];
}

// ---------------- temporal shift + layernorm on CLS rows ----------------
__global__ __launch_bounds__(256) void shift_ln_kernel(const float* __restrict__ x,
    const float* __restrict__ w1, const float* __restrict__ gam,
    const float* __restrict__ bet, float* __restrict__ t_out, float* __restrict__ s_out) {
  int i = blockIdx.x;                 // row 0..511 (= b*16 + seg)
  int seg = i & 15;
  const float* row  = x + (size_t)i * ROW_STRIDE;          // token 0 of seq i
  const float* prev = x + (size_t)(i - 1) * ROW_STRIDE;    // token 0 of seq i-1
  __shared__ float red[256];
  float vals[3];
  float lsum = 0.f;
  #pragma unroll
  for (int j = 0; j < 3; ++j) {
    int c = threadIdx.x + j * 256;
    float v = row[c];
    if (seg != 0) v += prev[c] * w1[c];
    vals[j] = v;
    lsum += v;
  }
  red[threadIdx.x] = lsum;
  __syncthreads();
  for (int off = 128; off > 0; off >>= 1) {
    if (threadIdx.x < off) red[threadIdx.x] += red[threadIdx.x + off];
    __syncthreads();
  }
  float mu = red[0] * (1.f / 768.f);
  __syncthreads();
  float lvar = 0.f;
  #pragma unroll
  for (int j = 0; j < 3; ++j) { float d = vals[j] - mu; lvar += d * d; }
  red[threadIdx.x] = lvar;
  __syncthreads();
  for (int off = 128; off > 0; off >>= 1) {
    if (threadIdx.x < off) red[threadIdx.x] += red[threadIdx.x + off];
    __syncthreads();
  }
  float rstd = rsqrtf(red[0] * (1.f / 768.f) + LN_EPS);
  #pragma unroll
  for (int j = 0; j < 3; ++j) {
    int c = threadIdx.x + j * 256;
    s_out[i * SEQ_C + c] = vals[j];
    t_out[i * SEQ_C + c] = (vals[j] - mu) * rstd * gam[c] + bet[c];
  }
}

// ---------------- fp32 WMMA GEMM: C(512x768) = A(512x768) @ W(768x768) + bias --------
// mode 0: write to contiguous buffer `out` (512x768)
// mode 1: add residual `res` and scatter rows into global output (row stride 197*768)
__global__ __launch_bounds__(256) void wmma_gemm_kernel(const float* __restrict__ A,
    const float* __restrict__ W, const float* __restrict__ bias,
    const float* __restrict__ res, float* __restrict__ out, int mode) {
  int wid   = (blockIdx.x << 3) + (threadIdx.x >> 5);  // 48 blocks * 8 waves = 384 waves
  int mTile = wid / 12;                                // 0..31  (16 rows each)
  int nStrip = wid % 12;                               // 0..11  (64 cols each)
  int lane  = threadIdx.x & 31;
  int nIdx  = lane & 15;                               // M index for A frag / N index for B,C frags
  int g2    = (lane >> 4) << 1;                        // K sub-offset: 0 (lanes 0-15) or 2 (16-31)
  const float* Arow = A + (size_t)(mTile * 16 + nIdx) * SEQ_C;
  int ncol0 = nStrip * 64 + nIdx;

  v8f acc[4] = {};
  for (int k0 = 0; k0 < SEQ_C; k0 += 4) {
    int ka = k0 + g2;
    v2f a;
    a.x = Arow[ka];
    a.y = Arow[ka + 1];
    const float* w0 = W + (size_t)ka * SEQ_C + ncol0;
    #pragma unroll
    for (int s = 0; s < 4; ++s) {
      v2f b;
      b.x = w0[s * 16];            // K = ka,   N = ncol0 + 16*s
      b.y = w0[SEQ_C + s * 16];    // K = ka+1, N = ncol0 + 16*s
      acc[s] = __builtin_amdgcn_wmma_f32_16x16x4_f32(
          false, a, false, b, (short)0, acc[s], false, false);
    }
  }

  int gOff = (lane >> 4) << 3;     // C/D: VGPR r holds M = r + 8*(lane>>4)
  #pragma unroll
  for (int s = 0; s < 4; ++s) {
    int ncol = ncol0 + s * 16;
    float bv = bias[ncol];
    #pragma unroll
    for (int r = 0; r < 8; ++r) {
      int mrow = mTile * 16 + r + gOff;
      float val = acc[s][r] + bv;
      if (mode) {
        val += res[mrow * SEQ_C + ncol];
        out[(size_t)mrow * ROW_STRIDE + ncol] = val;   // token-0 row of sequence mrow
      } else {
        out[mrow * SEQ_C + ncol] = val;
      }
    }
  }
}

// ---------------- tiny attention: one block per (batch b, head h), T=16, D=64 --------
__global__ __launch_bounds__(256) void attention_kernel(const float* __restrict__ q,
    const float* __restrict__ k, const float* __restrict__ v, float* __restrict__ ctx) {
  __shared__ float qs[16][64], ks[16][64], vs[16][64], ps[16][16];
  int b = blockIdx.x / 12;
  int h = blockIdx.x % 12;
  int base = (b * 16) * SEQ_C + h * 64;
  #pragma unroll
  for (int r = 0; r < 4; ++r) {
    int e  = r * 256 + threadIdx.x;   // 0..1023
    int ti = e >> 6, d = e & 63;
    int src = base + ti * SEQ_C + d;
    qs[ti][d] = q[src]; ks[ti][d] = k[src]; vs[ti][d] = v[src];
  }
  __syncthreads();
  // scores: thread (ti,si); rows of 16 threads sit in lanes [0..15] / [16..31] of a wave
  int ti = threadIdx.x >> 4, si = threadIdx.x & 15;
  float sc = 0.f;
  #pragma unroll
  for (int d = 0; d < 64; ++d) sc += qs[ti][d] * ks[si][d];
  sc *= 0.125f;                       // 1/sqrt(64)
  float m = sc;
  for (int off = 8; off > 0; off >>= 1) m = fmaxf(m, __shfl_xor(m, off, 16));
  float e = __expf(sc - m);
  float sum = e;
  for (int off = 8; off > 0; off >>= 1) sum += __shfl_xor(sum, off, 16);
  ps[ti][si] = e / sum;
  __syncthreads();
  // ctx[ti][d] = sum_s p[ti][s] * v[s][d]; 4 outputs per thread
  int d0 = si * 4;
  #pragma unroll
  for (int dd = 0; dd < 4; ++dd) {
    int d = d0 + dd;
    float acc = 0.f;
    #pragma unroll
    for (int s = 0; s < 16; ++s) acc += ps[ti][s] * vs[s][d];
    ctx[base + ti * SEQ_C + d] = acc;
  }
}

extern "C" void kernel_launch(void* const* d_in, const int* in_sizes, int n_in,
                              void* d_out, int out_size, void* d_ws, size_t ws_size,
                              hipStream_t stream) {
  const float* x  = (const float*)d_in[0];
  const float* w1 = (const float*)d_in[1];
  const float* lg = (const float*)d_in[2];
  const float* lb = (const float*)d_in[3];
  const float* Wq = (const float*)d_in[4];
  const float* bq = (const float*)d_in[5];
  const float* Wk = (const float*)d_in[6];
  const float* bk = (const float*)d_in[7];
  const float* Wv = (const float*)d_in[8];
  const float* bv = (const float*)d_in[9];
  const float* Wo = (const float*)d_in[10];
  const float* bo = (const float*)d_in[11];
  float* out = (float*)d_out;

  const size_t R = (size_t)NROWS * SEQ_C;     // 512*768 floats
  float* t   = (float*)d_ws;
  float* ss  = t  + R;
  float* q   = ss + R;
  float* kk  = q  + R;
  float* vv  = kk + R;
  float* ctx = vv + R;                        // total ~9.4 MB of d_ws

  int n4 = (NROWS * 197 * SEQ_C) / 4;         // float4 count of full tensor
  copy_kernel<<<(n4 + 255) / 256, 256, 0, stream>>>((const float4*)x, (float4*)out, n4);
  shift_ln_kernel<<<NROWS, 256, 0, stream>>>(x, w1, lg, lb, t, ss);
  wmma_gemm_kernel<<<48, 256, 0, stream>>>(t, Wq, bq, nullptr, q, 0);
  wmma_gemm_kernel<<<48, 256, 0, stream>>>(t, Wk, bk, nullptr, kk, 0);
  wmma_gemm_kernel<<<48, 256, 0, stream>>>(t, Wv, bv, nullptr, vv, 0);
  attention_kernel<<<384, 256, 0, stream>>>(q, kk, vv, ctx);
  wmma_gemm_kernel<<<48, 256, 0, stream>>>(ctx, Wo, bo, ss, out, 1);
}